// DAGNN_14491219657221
// MI455X (gfx1250) — compile-verified
//
#include <hip/hip_runtime.h>
#include <cstdint>
#include <cstddef>

#define DIN   256
#define DH    256
#define DOUT  64
#define KHOPS 10
#define PPW   704          // (KHOPS+1)*DOUT floats per node
#define PPW4  176          // PPW / 4 (float4 units)
#define SXP   260          // padded LDS row stride (bank-conflict-free)

typedef float v2f __attribute__((ext_vector_type(2)));
typedef float v8f __attribute__((ext_vector_type(8)));

// ---------------- degree / normalization prep ----------------

__global__ void k_deg_init(float* __restrict__ dinv, int N) {
    int i = blockIdx.x * blockDim.x + threadIdx.x;
    if (i < N) dinv[i] = 1.0f;                      // self-loop contributes 1
}

__global__ void k_deg_count(const long long* __restrict__ e64,
                            float* __restrict__ dinv, int E) {
    int e = blockIdx.x * blockDim.x + threadIdx.x;
    if (e < E) atomicAdd(&dinv[(int)e64[(size_t)E + e]], 1.0f);
}

__global__ void k_rsqrt(float* __restrict__ dinv, int N) {
    int i = blockIdx.x * blockDim.x + threadIdx.x;
    if (i < N) dinv[i] = rsqrtf(dinv[i]);           // deg >= 1 always
}

__global__ void k_edge_prep(const long long* __restrict__ e64,
                            const float* __restrict__ dinv,
                            int* __restrict__ rows, int* __restrict__ cols,
                            float* __restrict__ nrm, int E) {
    int e = blockIdx.x * blockDim.x + threadIdx.x;
    if (e >= E) return;
    int r = (int)e64[e];
    int c = (int)e64[(size_t)E + e];
    rows[e] = r;
    cols[e] = c;
    nrm[e]  = dinv[r] * dinv[c];
}

// ---------------- fused MLP: relu(x@W1+b1)@W2+b2 -> pps[:,0,:] ----------------
// One block = one 16-row strip. 16 waves: phase 1 each wave does a 16x16 tile of
// h1 (cols w*16..) via V_WMMA_F32_16X16X4_F32, ReLU into LDS; phase 2 waves 0..3
// do the 16x64 second GEMM and store hop-0 features.

__global__ __launch_bounds__(512)
void k_mlp_wmma(const float* __restrict__ x,  const float* __restrict__ W1,
                const float* __restrict__ b1, const float* __restrict__ W2,
                const float* __restrict__ b2, float* __restrict__ pps) {
    __shared__ float sx[16 * SXP];
    __shared__ float sh[16 * SXP];

    const int tid  = threadIdx.x;
    const int wave = tid >> 5;
    const int lane = tid & 31;
    const int m    = lane & 15;        // row (A/D) or col (B/D) index
    const int hi   = lane >> 4;        // selects K pair {0,1} vs {2,3}
    const long long r0 = (long long)blockIdx.x * 16;

    // Load x strip [16 x 256] into LDS (vectorized float4, padded rows).
    {
        const float4* xg = (const float4*)(x + (size_t)r0 * DIN);
        for (int i = tid; i < 16 * 64; i += 512) {
            int rr = i >> 6, cc = i & 63;
            float4 v = xg[rr * 64 + cc];
            float* d = &sx[rr * SXP + cc * 4];
            d[0] = v.x; d[1] = v.y; d[2] = v.z; d[3] = v.w;
        }
    }
    __syncthreads();

    // Phase 1: h1 tile (16 x 16) per wave, K = 256 in steps of 4.
    {
        const int colb = wave * 16;
        v8f c = {};
        for (int k0 = 0; k0 < DIN; k0 += 4) {
            const int ka = k0 + 2 * hi;
            v2f a = *(const v2f*)&sx[m * SXP + ka];       // ds_load_b64
            v2f b;
            b.x = W1[(size_t)ka       * DH + colb + m];
            b.y = W1[(size_t)(ka + 1) * DH + colb + m];
            c = __builtin_amdgcn_wmma_f32_16x16x4_f32(
                    false, a, false, b, (short)0, c, false, false);
        }
        const float bias = b1[colb + m];
#pragma unroll
        for (int i = 0; i < 8; ++i) {
            float v = c[i] + bias;
            v = v > 0.0f ? v : 0.0f;                      // ReLU
            sh[(i + 8 * hi) * SXP + colb + m] = v;
        }
    }
    __syncthreads();

    // Phase 2: h = h1 @ W2 + b2 (16 x 64), waves 0..3, write hop-0 slice.
    if (wave < 4) {
        const int colb = wave * 16;
        v8f c = {};
        for (int k0 = 0; k0 < DH; k0 += 4) {
            const int ka = k0 + 2 * hi;
            v2f a = *(const v2f*)&sh[m * SXP + ka];
            v2f b;
            b.x = W2[(size_t)ka       * DOUT + colb + m];
            b.y = W2[(size_t)(ka + 1) * DOUT + colb + m];
            c = __builtin_amdgcn_wmma_f32_16x16x4_f32(
                    false, a, false, b, (short)0, c, false, false);
        }
        const float bias = b2[colb + m];
#pragma unroll
        for (int i = 0; i < 8; ++i) {
            const long long row = r0 + i + 8 * hi;
            pps[(size_t)row * PPW + colb + m] = c[i] + bias;
        }
    }
}

// ---------------- hop propagation ----------------

// self-loop term + slice init: pps[:,k,:] = dinv^2 * pps[:,k-1,:]
__global__ void k_hop_init(const float* __restrict__ dinv,
                           float* __restrict__ pps, int N, int k) {
    int idx = blockIdx.x * blockDim.x + threadIdx.x;
    if (idx >= N * 16) return;
    int n = idx >> 4, q = idx & 15;
    float d = dinv[n];
    float s = d * d;
    float4 v = ((const float4*)pps)[(size_t)n * PPW4 + (size_t)(k - 1) * 16 + q];
    ((float4*)pps)[(size_t)n * PPW4 + (size_t)k * 16 + q] =
        make_float4(s * v.x, s * v.y, s * v.z, s * v.w);
}

// edge scatter: pps[col,k,:] += norm_e * pps[row,k-1,:]   (f32 atomics, L2-hot)
__global__ void k_hop_scatter(const int* __restrict__ rows,
                              const int* __restrict__ cols,
                              const float* __restrict__ nrm,
                              float* __restrict__ pps, int E, int k) {
    size_t idx = (size_t)blockIdx.x * blockDim.x + threadIdx.x;
    if (idx >= (size_t)E * 16) return;
    int e = (int)(idx >> 4), q = (int)(idx & 15);
    int r = rows[e], c = cols[e];
    float w = nrm[e];
    float4 v = ((const float4*)pps)[(size_t)r * PPW4 + (size_t)(k - 1) * 16 + q];
    float* dst = pps + (size_t)c * PPW + (size_t)k * DOUT + q * 4;
    atomicAdd(dst + 0, w * v.x);
    atomicAdd(dst + 1, w * v.y);
    atomicAdd(dst + 2, w * v.z);
    atomicAdd(dst + 3, w * v.w);
}

// ---------------- adaptive hop pooling ----------------
// One wave32 per node; each lane owns 2 channels; shfl_xor reduction for the
// score dot-product.

__global__ void k_pool(const float* __restrict__ pps,
                       const float* __restrict__ proj_w,
                       const float* __restrict__ proj_b,
                       float* __restrict__ out, int N) {
    int gw   = (int)((blockIdx.x * blockDim.x + threadIdx.x) >> 5);
    int lane = threadIdx.x & 31;
    if (gw >= N) return;                               // wave-uniform exit

    const float2 pw = ((const float2*)proj_w)[lane];
    const float  pb = proj_b[0];
    const float2* row = (const float2*)(pps + (size_t)gw * PPW);

    float a0 = 0.0f, a1 = 0.0f;
#pragma unroll
    for (int k = 0; k <= KHOPS; ++k) {
        float2 v = row[k * 32 + lane];
        float p = v.x * pw.x + v.y * pw.y;
#pragma unroll
        for (int off = 16; off > 0; off >>= 1)
            p += __shfl_xor(p, off, 32);
        float s = 1.0f / (1.0f + __expf(-(p + pb)));
        a0 += s * v.x;
        a1 += s * v.y;
    }
    ((float2*)out)[(size_t)gw * 32 + lane] = make_float2(a0, a1);
}

// ---------------- launcher ----------------

extern "C" void kernel_launch(void* const* d_in, const int* in_sizes, int n_in,
                              void* d_out, int out_size, void* d_ws, size_t ws_size,
                              hipStream_t stream) {
    const float*     x   = (const float*)d_in[0];
    const long long* e64 = (const long long*)d_in[1];   // int64 edge_index [2,E]
    const float*     W1  = (const float*)d_in[2];
    const float*     b1  = (const float*)d_in[3];
    const float*     W2  = (const float*)d_in[4];
    const float*     b2  = (const float*)d_in[5];
    const float*     pw  = (const float*)d_in[6];
    const float*     pb  = (const float*)d_in[7];
    float*           out = (float*)d_out;

    const int N = in_sizes[0] / DIN;   // 100000
    const int E = in_sizes[1] / 2;     // 3200000

    // workspace layout
    float* pps  = (float*)d_ws;                         // N * 704 floats
    float* dinv = pps + (size_t)N * PPW;                // N floats (deg -> dinv in place)
    int*   rows = (int*)(dinv + N);                     // E ints
    int*   cols = rows + E;                             // E ints
    float* nrm  = (float*)(cols + E);                   // E floats
    (void)n_in; (void)out_size; (void)ws_size;

    k_deg_init <<<(N + 255) / 256, 256, 0, stream>>>(dinv, N);
    k_deg_count<<<(E + 255) / 256, 256, 0, stream>>>(e64, dinv, E);
    k_rsqrt    <<<(N + 255) / 256, 256, 0, stream>>>(dinv, N);
    k_edge_prep<<<(E + 255) / 256, 256, 0, stream>>>(e64, dinv, rows, cols, nrm, E);

    k_mlp_wmma<<<N / 16, 512, 0, stream>>>(x, W1, b1, W2, b2, pps);

    for (int k = 1; k <= KHOPS; ++k) {
        k_hop_init<<<((size_t)N * 16 + 255) / 256, 256, 0, stream>>>(dinv, pps, N, k);
        size_t tot = (size_t)E * 16;
        k_hop_scatter<<<(unsigned)((tot + 255) / 256), 256, 0, stream>>>(
            rows, cols, nrm, pps, E, k);
    }

    k_pool<<<((size_t)N * 32 + 255) / 256, 256, 0, stream>>>(pps, pw, pb, out, N);
}